// QLayer_65481071396233
// MI455X (gfx1250) — compile-verified
//
#include <hip/hip_runtime.h>
#include <math.h>

// out[b,q] = prod_{j<=q} cos(x[b,j] + params[j])   for b in [0,512), q in [0,15)
//
// Implemented as: la = log|cos|, ng = (cos<0), then per 16-row batch tile
//   Lsum = A_la (16x16) x LowerTriOnes (16x16)   via 4 chained V_WMMA_F32_16X16X4_F32
//   Psum = A_ng (16x16) x LowerTriOnes (16x16)   via 4 chained V_WMMA_F32_16X16X4_F32
//   out  = (-1)^{(int)Psum & 1} * exp(Lsum)
// One wave32 per tile; 512/16 = 32 tiles -> grid(32), block(32). EXEC all-ones (WMMA req).

typedef __attribute__((ext_vector_type(2))) float v2f;
typedef __attribute__((ext_vector_type(8))) float v8f;

#define NQ    15
#define BATCH 512

__global__ __launch_bounds__(32) void qlayer_cumcos_wmma(const float* __restrict__ x,
                                                         const float* __restrict__ params,
                                                         float* __restrict__ out)
{
    const int lane  = threadIdx.x;     // 0..31
    const int tile  = blockIdx.x;      // 0..31, 16 batch rows each
    const int m     = lane & 15;       // batch row within tile (A-matrix M, also B/D column N)
    const int row   = tile * 16 + m;   // global batch row this lane's A data belongs to
    const int hi    = lane >> 4;       // 0: lanes 0-15, 1: lanes 16-31
    const int kbase = hi * 2;          // A/B K sub-offset within each K=4 chunk

    // For WMMA chunk k (K = 4k..4k+3), this lane supplies A[m][4k+kbase+{0,1}].
    // Precompute log|cos(theta_j)| and the negative-sign indicator for those 8 j's.
    float la[8];
    float ng[8];
#pragma unroll
    for (int k = 0; k < 4; ++k) {
#pragma unroll
        for (int c = 0; c < 2; ++c) {
            const int j = 4 * k + kbase + c;
            float lv = 0.0f, nv = 0.0f;
            if (j < NQ) {
                const float th  = x[row * NQ + j] + params[j];
                const float cth = __cosf(th);
                nv = (cth < 0.0f) ? 1.0f : 0.0f;
                lv = __logf(fabsf(cth));   // cth==0 -> -inf -> exp -> 0 (correct cumprod)
            }
            la[2 * k + c] = lv;            // K=15 padded with 0
            ng[2 * k + c] = nv;
        }
    }

    // Two chained WMMA accumulations over K: logs and sign-parity counts.
    v8f accL = {};
    v8f accP = {};
#pragma unroll
    for (int k = 0; k < 4; ++k) {
        v2f a, p, b;
        a.x = la[2 * k];  a.y = la[2 * k + 1];
        p.x = ng[2 * k];  p.y = ng[2 * k + 1];
        // B chunk: B[j][q] = (j <= q) ? 1 : 0, this lane supplies rows j0, j0+1 of column q=m.
        const int j0 = 4 * k + kbase;
        b.x = (j0     <= m) ? 1.0f : 0.0f;
        b.y = (j0 + 1 <= m) ? 1.0f : 0.0f;
        // 8 args: (neg_a, A, neg_b, B, c_mod, C, reuse_a, reuse_b)
        accL = __builtin_amdgcn_wmma_f32_16x16x4_f32(false, a, false, b, (short)0, accL, false, false);
        accP = __builtin_amdgcn_wmma_f32_16x16x4_f32(false, p, false, b, (short)0, accP, false, false);
    }

    // D layout: VGPR v, lane l -> M = v + 8*(l>=16), N = l&15.
    const int N = m;
    if (N < NQ) {
#pragma unroll
        for (int v = 0; v < 8; ++v) {
            const int M   = v + 8 * hi;
            const int par = ((int)(accP[v] + 0.5f)) & 1;   // exact small-int sum in f32
            const float mag = __expf(accL[v]);
            out[(tile * 16 + M) * NQ + N] = par ? -mag : mag;
        }
    }
}

extern "C" void kernel_launch(void* const* d_in, const int* in_sizes, int n_in,
                              void* d_out, int out_size, void* d_ws, size_t ws_size,
                              hipStream_t stream) {
    (void)in_sizes; (void)n_in; (void)d_ws; (void)ws_size; (void)out_size;
    const float* x      = (const float*)d_in[0];   // (512, 15) f32
    const float* params = (const float*)d_in[1];   // (15,)    f32
    float*       out    = (float*)d_out;           // (512, 15) f32

    qlayer_cumcos_wmma<<<BATCH / 16, 32, 0, stream>>>(x, params, out);
}